// TGCNCholeskyModel_34239479284353
// MI455X (gfx1250) — compile-verified
//
#include <hip/hip_runtime.h>
#include <math.h>

// ---- problem sizes (fixed by the reference) ----
constexpr int T_ = 8, N_ = 1024, C_ = 64, H_ = 256, E_ = 32768;
constexpr long NC_ = (long)N_ * (N_ + 1) / 2;   // 524800
#define CLIPV 100000.0f

typedef float v2f __attribute__((ext_vector_type(2)));
typedef float v8f __attribute__((ext_vector_type(8)));

__device__ __forceinline__ float ntn(float v) {
    if (__builtin_isnan(v)) return 0.f;
    if (__builtin_isinf(v)) return v > 0.f ? CLIPV : -CLIPV;
    return v;
}
__device__ __forceinline__ float sigf(float x) { return 1.f / (1.f + expf(-x)); }

// ------------------------------------------------------------------
// Batched WMMA GEMM: C[b] = A[b] @ B (row-major fp32).
// Block = 8 wave32 -> one 16(M) x 128(N) C panel; the 16-row A chunk is
// shared by all 8 waves, staged in LDS via GLOBAL_LOAD_ASYNC_TO_LDS_B128
// with double buffering (ASYNCcnt), consumed as ds_load_b64 A-fragments
// for V_WMMA_F32_16X16X4_F32.  mode 1: nan_to_num on store.
// ------------------------------------------------------------------
__global__ void k_wmma_gemm(const float* __restrict__ A, int lda, long strideA,
                            const float* __restrict__ B, int ldb,
                            float* __restrict__ C, int ldc, long strideC,
                            int K, int mode)
{
    constexpr int KT = 32;                       // K-chunk staged in LDS
    __shared__ float sA[2][16 * KT];             // 2 x 2KB double buffer

    const int lane = threadIdx.x;                // 0..31
    const int wave = threadIdx.y;                // 0..7
    const int hl   = lane >> 4;                  // lane half
    const int l16  = lane & 15;
    const int kb   = hl * 2;                     // lanes 0-15: K=0,1 ; 16-31: K=2,3
    const int tid  = wave * 32 + lane;

    const int ntile = blockIdx.x * 8 + wave;
    const int mtile = blockIdx.y;
    const int b     = blockIdx.z;

    const float* Ab = A + (long)b * strideA + (long)(mtile * 16) * lda;
    float*       Cb = C + (long)b * strideC;
    const int ncol0 = ntile * 16;

    // producer mapping (waves 0-3 only, uniform per wave): 128 lanes x 16B
    const int arow = tid >> 3;                   // 0..15
    const int acol = (tid & 7) * 4;              // 0,4,...,28

    v8f acc = {0.f, 0.f, 0.f, 0.f, 0.f, 0.f, 0.f, 0.f};

    // prime chunk 0
    if (tid < 128) {
        unsigned lds = (unsigned)(size_t)&sA[0][arow * KT + acol];
        const float* g = Ab + (long)arow * lda + acol;
        asm volatile("global_load_async_to_lds_b128 %0, %1, off"
                     :: "v"(lds), "v"(g) : "memory");
    }

    const int NK = K / KT;
    for (int kc = 0; kc < NK; ++kc) {
        const int k0 = kc * KT;
        if (kc + 1 < NK && tid < 128) {          // issue next chunk
            unsigned lds = (unsigned)(size_t)&sA[(kc + 1) & 1][arow * KT + acol];
            const float* g = Ab + (long)arow * lda + (k0 + KT + acol);
            asm volatile("global_load_async_to_lds_b128 %0, %1, off"
                         :: "v"(lds), "v"(g) : "memory");
        }
        if (tid < 128) {                         // async loads complete in order
            if (kc + 1 < NK) asm volatile("s_wait_asynccnt 0x1" ::: "memory");
            else             asm volatile("s_wait_asynccnt 0x0" ::: "memory");
        }
        __syncthreads();

        const float* As = &sA[kc & 1][0];
#pragma unroll
        for (int kk = 0; kk < KT; kk += 4) {
            v2f av = *(const v2f*)&As[l16 * KT + kk + kb];   // ds_load_b64
            v2f bv;
            bv.x = B[(long)(k0 + kk + kb)     * ldb + (ncol0 + l16)];
            bv.y = B[(long)(k0 + kk + kb + 1) * ldb + (ncol0 + l16)];
            acc = __builtin_amdgcn_wmma_f32_16x16x4_f32(
                false, av, false, bv, (short)0, acc, false, false);
        }
        __syncthreads();                         // protect buffer reuse
    }

#pragma unroll
    for (int i = 0; i < 8; ++i) {
        int row = mtile * 16 + i + 8 * hl;
        float v = acc[i];
        if (mode == 1) v = ntn(v);
        Cb[(long)row * ldc + (ncol0 + l16)] = v;
    }
}

// ---------------- GCN degree / normalization ----------------
__global__ void k_deg_init(float* deg) {
    int i = blockIdx.x * blockDim.x + threadIdx.x;
    if (i < T_ * N_) deg[i] = 1.0f;                       // self-loop weight
}
__global__ void k_deg_scatter(const int* __restrict__ ei,
                              const float* __restrict__ ew, float* deg) {
    int i = blockIdx.x * blockDim.x + threadIdx.x;
    if (i >= T_ * E_) return;
    int e = i % E_, t = i / E_;
    int col = ei[(long)t * 2 * E_ + E_ + e];
    atomicAdd(&deg[t * N_ + col], ew[i]);
}
__global__ void k_deg_fin(float* deg) {
    int i = blockIdx.x * blockDim.x + threadIdx.x;
    if (i < T_ * N_) deg[i] = rsqrtf(deg[i]);             // deg >= 1 always
}

// ---------------- GCN aggregation ----------------
__global__ void k_agg_self(const float* __restrict__ h, float* __restrict__ out,
                           const float* __restrict__ dinv) {
    long i = (long)blockIdx.x * blockDim.x + threadIdx.x;
    if (i >= (long)T_ * N_ * H_) return;
    long rem = i / H_;
    int n = rem % N_, t = rem / N_;
    float d = dinv[t * N_ + n];
    out[i] = d * d * h[i];
}
__global__ void k_agg_scatter(const float* __restrict__ h, float* __restrict__ out,
                              const int* __restrict__ ei, const float* __restrict__ ew,
                              const float* __restrict__ dinv) {
    long i = (long)blockIdx.x * blockDim.x + threadIdx.x;
    if (i >= (long)T_ * E_ * H_) return;
    int f = i % H_;
    long rem = i / H_;
    int e = rem % E_, t = rem / E_;
    int row = ei[(long)t * 2 * E_ + e];
    int col = ei[(long)t * 2 * E_ + E_ + e];
    float nrm = dinv[t * N_ + row] * ew[(long)t * E_ + e] * dinv[t * N_ + col];
    atomicAdd(&out[((long)t * N_ + col) * H_ + f],
              nrm * h[((long)t * N_ + row) * H_ + f]);
}
__global__ void k_finalize_relu(float* out, const float* __restrict__ bias) {
    long i = (long)blockIdx.x * blockDim.x + threadIdx.x;
    if (i >= (long)T_ * N_ * H_) return;
    float v = ntn(out[i] + bias[i % H_]);
    out[i] = v > 0.f ? v : 0.f;
}

// ---------------- global mean pool ----------------
__global__ void k_mean(const float* __restrict__ h, float* embs) {
    int i = blockIdx.x * blockDim.x + threadIdx.x;
    if (i >= T_ * H_) return;
    int f = i % H_, t = i / H_;
    float s = 0.f;
    for (int n = 0; n < N_; ++n) s += h[((long)t * N_ + n) * H_ + f];
    embs[i] = ntn(ntn(s * (1.0f / N_)));
}

// ---------------- LSTM (single block, 1024 gates) ----------------
__global__ void k_lstm(const float* __restrict__ embs,
                       const float* __restrict__ w_ih, const float* __restrict__ w_hh,
                       const float* __restrict__ b_ih, const float* __restrict__ b_hh,
                       float* final_h) {
    __shared__ float sx[H_], sh[H_], sc[H_], sg[4 * H_];
    int j = threadIdx.x;                    // 0..1023
    if (j < H_) { sh[j] = 0.f; sc[j] = 0.f; }
    __syncthreads();
    for (int t = 0; t < T_; ++t) {
        if (j < H_) sx[j] = embs[t * H_ + j];
        __syncthreads();
        float g = b_ih[j] + b_hh[j];
        for (int k = 0; k < H_; ++k)
            g += sx[k] * w_ih[(long)j * H_ + k] + sh[k] * w_hh[(long)j * H_ + k];
        sg[j] = g;
        __syncthreads();
        if (j < H_) {
            float ig = sigf(sg[j]);
            float fg = sigf(sg[H_ + j]);
            float gg = tanhf(sg[2 * H_ + j]);
            float og = sigf(sg[3 * H_ + j]);
            float c  = fg * sc[j] + ig * gg;
            sc[j] = c;
            sh[j] = og * tanhf(c);
        }
        __syncthreads();
    }
    if (j < H_) final_h[j] = ntn(ntn(sh[j]));
}

// ---------------- fc GEMV: the HBM-bound phase (537 MB stream) ----------------
__global__ void k_fc(const float* __restrict__ fh, const float* __restrict__ fc_w,
                     const float* __restrict__ fc_b, float* vec) {
    __shared__ float s[H_];
    int tid = threadIdx.x;
    s[tid] = fh[tid];
    __syncthreads();
    long j = (long)blockIdx.x * 256 + tid;         // NC_ = 2050*256 exactly
    float acc = fc_b[j];
    for (int k0 = 0; k0 < H_; k0 += 64) {
        __builtin_prefetch(&fc_w[(long)(k0 + 64) * NC_ + j], 0, 0);  // global_prefetch_b8
#pragma unroll 8
        for (int k = k0; k < k0 + 64; ++k)
            acc += s[k] * fc_w[(long)k * NC_ + j];
    }
    vec[j] = ntn(acc);
}

// ---------------- scatter vec -> L and L^T ----------------
__global__ void k_build_L(const float* __restrict__ vec, float* L, float* LT) {
    long idx = (long)blockIdx.x * blockDim.x + threadIdx.x;
    if (idx >= (long)N_ * N_) return;
    int jj = idx % N_, i = idx / N_;
    float v = (jj <= i) ? vec[(long)i * (i + 1) / 2 + jj] : 0.f;
    L[(long)i * N_ + jj] = v;
    LT[(long)jj * N_ + i] = v;
}

extern "C" void kernel_launch(void* const* d_in, const int* in_sizes, int n_in,
                              void* d_out, int out_size, void* d_ws, size_t ws_size,
                              hipStream_t stream) {
    const float* xs   = (const float*)d_in[0];
    const int*   ei   = (const int*)  d_in[1];
    const float* ew   = (const float*)d_in[2];
    const float* w1   = (const float*)d_in[3];
    const float* b1   = (const float*)d_in[4];
    const float* w2   = (const float*)d_in[5];
    const float* b2   = (const float*)d_in[6];
    const float* w_ih = (const float*)d_in[7];
    const float* w_hh = (const float*)d_in[8];
    const float* b_ih = (const float*)d_in[9];
    const float* b_hh = (const float*)d_in[10];
    const float* fc_w = (const float*)d_in[11];
    const float* fc_b = (const float*)d_in[12];
    float* out = (float*)d_out;

    float* ws   = (float*)d_ws;
    float* bufA = ws;  ws += (long)T_ * N_ * H_;
    float* bufB = ws;  ws += (long)T_ * N_ * H_;
    float* dinv = ws;  ws += (long)T_ * N_;
    float* embs = ws;  ws += T_ * H_;
    float* fh   = ws;  ws += H_;
    float* vec  = ws;  ws += NC_;
    float* L    = ws;  ws += (long)N_ * N_;
    float* LT   = ws;  ws += (long)N_ * N_;

    dim3 blk(256);
    dim3 wblk(32, 8);                                  // 8 wave32 per block
    const long tnh = (long)T_ * N_ * H_;
    const long teh = (long)T_ * E_ * H_;

    // symmetric normalization
    k_deg_init   <<<(T_ * N_ + 255) / 256, blk, 0, stream>>>(dinv);
    k_deg_scatter<<<(T_ * E_ + 255) / 256, blk, 0, stream>>>(ei, ew, dinv);
    k_deg_fin    <<<(T_ * N_ + 255) / 256, blk, 0, stream>>>(dinv);

    // conv1: bufA = xs @ w1   (M=1024, K=64, N=256, batch=8)
    k_wmma_gemm<<<dim3(H_ / 16 / 8, N_ / 16, T_), wblk, 0, stream>>>(
        xs, C_, (long)N_ * C_, w1, H_, bufA, H_, (long)N_ * H_, C_, 0);
    k_agg_self   <<<(tnh + 255) / 256, blk, 0, stream>>>(bufA, bufB, dinv);
    k_agg_scatter<<<(teh + 255) / 256, blk, 0, stream>>>(bufA, bufB, ei, ew, dinv);
    k_finalize_relu<<<(tnh + 255) / 256, blk, 0, stream>>>(bufB, b1);

    // conv2: bufA = bufB @ w2  (K=256)
    k_wmma_gemm<<<dim3(H_ / 16 / 8, N_ / 16, T_), wblk, 0, stream>>>(
        bufB, H_, (long)N_ * H_, w2, H_, bufA, H_, (long)N_ * H_, H_, 0);
    k_agg_self   <<<(tnh + 255) / 256, blk, 0, stream>>>(bufA, bufB, dinv);
    k_agg_scatter<<<(teh + 255) / 256, blk, 0, stream>>>(bufA, bufB, ei, ew, dinv);
    k_finalize_relu<<<(tnh + 255) / 256, blk, 0, stream>>>(bufB, b2);

    // temporal head
    k_mean<<<(T_ * H_ + 255) / 256, blk, 0, stream>>>(bufB, embs);
    k_lstm<<<1, 1024, 0, stream>>>(embs, w_ih, w_hh, b_ih, b_hh, fh);
    k_fc  <<<(int)(NC_ / 256), blk, 0, stream>>>(fh, fc_w, fc_b, vec);

    // Cholesky-style outer product: out = L @ L^T (fp32 WMMA, K=1024)
    k_build_L<<<(int)(((long)N_ * N_ + 255) / 256), blk, 0, stream>>>(vec, L, LT);
    k_wmma_gemm<<<dim3(N_ / 16 / 8, N_ / 16, 1), wblk, 0, stream>>>(
        L, N_, 0, LT, N_, out, N_, 0, N_, 1);
}